// MambaBlock_42047729828541
// MI455X (gfx1250) — compile-verified
//
#include <hip/hip_runtime.h>
#include <math.h>

// ---------------- problem constants (from reference) ----------------
#define BATCH   4
#define CCH     64          // C
#define LSEQ    4096        // H*W
#define DM      128         // d_model
#define DI      256         // d_inner
#define DS      16          // d_state
#define DTR     8           // dt_rank
#define KCONV   4
#define NCHUNK  64          // scan chunks per sequence
#define CLEN    (LSEQ / NCHUNK)   // 64 tokens per chunk

typedef __attribute__((ext_vector_type(2))) float v2f;
typedef __attribute__((ext_vector_type(8))) float v8f;

// CDNA5 fp32 WMMA: D(16x16) = A(16x4) * B(4x16) + C
__device__ __forceinline__ v8f wmma4(v2f a, v2f b, v8f c) {
  return __builtin_amdgcn_wmma_f32_16x16x4_f32(
      /*neg_a=*/false, a, /*neg_b=*/false, b,
      /*c_mod=*/(short)0, c, /*reuse_a=*/false, /*reuse_b=*/false);
}

__device__ __forceinline__ float silu_f(float v) {
  return v / (1.0f + __expf(-v));
}

// =====================================================================
// k1: token proj (C->DM) + LayerNorm + in_proj (DM->2*DI); per-wave 16-token tile
// =====================================================================
#define XS_PAD 68    // 64 + 4   (stride % 64 == 4 -> conflict-free A reads)
#define XM_PAD 132   // 128 + 4
__global__ __launch_bounds__(64) void k1_proj_ln_inproj(
    const float* __restrict__ x, const float* __restrict__ proj_w,
    const float* __restrict__ proj_b, const float* __restrict__ ln_g,
    const float* __restrict__ ln_b, const float* __restrict__ in_proj_w,
    float* __restrict__ xi, float* __restrict__ z) {
  __shared__ float xs[2][16][XS_PAD];
  __shared__ float xm[2][16][XM_PAD];
  const int lane = threadIdx.x & 31;
  const int wv   = threadIdx.x >> 5;
  const int tile = blockIdx.x * 2 + wv;
  const int tok0 = tile * 16;
  const int b    = tok0 >> 12;            // / LSEQ
  const int l0   = tok0 & (LSEQ - 1);
  const int row  = lane & 15;             // A row == B/C column index
  const int half = lane >> 4;
  const int kb   = half * 2;

  // ---- stage 16x64 x tile (coalesced 16-token runs per channel) ----
  {
    const float* xb = x + ((size_t)b * CCH) * LSEQ + l0;
    #pragma unroll 4
    for (int ci = 0; ci < 32; ++ci) {
      int c = ci * 2 + half;
      xs[wv][row][c] = xb[(size_t)c * LSEQ + row];
    }
  }
  __syncthreads();

  // ---- xm = xs(16x64) @ proj_w(64x128) + proj_b ----
  for (int nt = 0; nt < DM / 16; ++nt) {
    int n = nt * 16 + row;
    v8f acc = {};
    #pragma unroll 4
    for (int kk = 0; kk < CCH / 4; ++kk) {
      int k = kk * 4 + kb;
      v2f a, bb;
      a.x  = xs[wv][row][k];
      a.y  = xs[wv][row][k + 1];
      bb.x = proj_w[k * DM + n];
      bb.y = proj_w[(k + 1) * DM + n];
      acc  = wmma4(a, bb, acc);
    }
    float bias = proj_b[n];
    #pragma unroll
    for (int i = 0; i < 8; ++i)
      xm[wv][i + half * 8][n] = acc[i] + bias;
  }
  __syncthreads();

  // ---- LayerNorm over 128 cols: 2 lanes per row, shfl_xor combine ----
  {
    float s = 0.f, s2 = 0.f;
    #pragma unroll 8
    for (int j = 0; j < 64; ++j) {
      float v = xm[wv][row][half * 64 + j];
      s += v; s2 += v * v;
    }
    s  += __shfl_xor(s, 16);
    s2 += __shfl_xor(s2, 16);
    float mu   = s * (1.0f / 128.0f);
    float var  = s2 * (1.0f / 128.0f) - mu * mu;
    float rstd = rsqrtf(var + 1e-5f);
    #pragma unroll 8
    for (int j = 0; j < 64; ++j) {
      int col = half * 64 + j;
      float v = xm[wv][row][col];
      xm[wv][row][col] = (v - mu) * rstd * ln_g[col] + ln_b[col];
    }
  }
  __syncthreads();

  // ---- xz = xm(16x128) @ in_proj_w(128x512); split xi | z ----
  for (int nt = 0; nt < (2 * DI) / 16; ++nt) {
    int n = nt * 16 + row;
    v8f acc = {};
    #pragma unroll 4
    for (int kk = 0; kk < DM / 4; ++kk) {
      int k = kk * 4 + kb;
      v2f a, bb;
      a.x  = xm[wv][row][k];
      a.y  = xm[wv][row][k + 1];
      bb.x = in_proj_w[k * (2 * DI) + n];
      bb.y = in_proj_w[(k + 1) * (2 * DI) + n];
      acc  = wmma4(a, bb, acc);
    }
    float* dst = (n < DI) ? xi : z;       // uniform per nt
    int col = n & (DI - 1);
    size_t base = ((size_t)b * LSEQ + l0) * DI + col;
    #pragma unroll
    for (int i = 0; i < 8; ++i)
      dst[base + (size_t)(i + half * 8) * DI] = acc[i];
  }
}

// =====================================================================
// k2: depthwise causal conv (K=4) + SiLU  -> u
// =====================================================================
__global__ __launch_bounds__(256) void k2_conv_silu(
    const float* __restrict__ xi, const float* __restrict__ conv_w,
    const float* __restrict__ conv_b, float* __restrict__ u) {
  int idx = blockIdx.x * 256 + threadIdx.x;     // B*L*DI threads
  int d = idx & (DI - 1);
  int l = (idx >> 8) & (LSEQ - 1);
  int b = idx >> 20;
  float acc = conv_b[d];
  #pragma unroll
  for (int k = 0; k < KCONV; ++k) {
    int ls = l - (KCONV - 1) + k;
    float xv = (ls >= 0) ? xi[(((size_t)b * LSEQ) + ls) * DI + d] : 0.f;
    acc = fmaf(conv_w[d * KCONV + k], xv, acc);
  }
  u[idx] = silu_f(acc);
}

// =====================================================================
// k3: x_proj (DI->40) then dt_proj (8->DI) + softplus; emits dt, B, C
// =====================================================================
#define US_PAD 260   // 256 + 4
#define XD_PAD 52    // 48  + 4
__global__ __launch_bounds__(64) void k3_xproj_dt(
    const float* __restrict__ u, const float* __restrict__ x_proj_w,
    const float* __restrict__ dt_proj_w, const float* __restrict__ dt_proj_b,
    float* __restrict__ dt, float* __restrict__ Bm, float* __restrict__ Cm) {
  __shared__ float us[2][16][US_PAD];
  __shared__ float xd[2][16][XD_PAD];
  const int lane = threadIdx.x & 31;
  const int wv   = threadIdx.x >> 5;
  const int tile = blockIdx.x * 2 + wv;
  const int tok0 = tile * 16;
  const int b    = tok0 >> 12;
  const int l0   = tok0 & (LSEQ - 1);
  const int row  = lane & 15;
  const int half = lane >> 4;
  const int kb   = half * 2;

  // stage u tile 16xDI (coalesced)
  #pragma unroll 4
  for (int it = 0; it < (16 * DI) / 32; ++it) {
    int idx = it * 32 + lane;
    int r = idx >> 8, k = idx & (DI - 1);
    us[wv][r][k] = u[((size_t)b * LSEQ + l0 + r) * DI + k];
  }
  __syncthreads();

  // x_dbl = u(16x256) @ x_proj_w(256x40)  (3 column tiles, padded)
  for (int nt = 0; nt < 3; ++nt) {
    int n = nt * 16 + row;
    bool ok = (n < DTR + 2 * DS);
    v8f acc = {};
    #pragma unroll 4
    for (int kk = 0; kk < DI / 4; ++kk) {
      int k = kk * 4 + kb;
      v2f a, bb;
      a.x  = us[wv][row][k];
      a.y  = us[wv][row][k + 1];
      bb.x = ok ? x_proj_w[k * 40 + n] : 0.f;
      bb.y = ok ? x_proj_w[(k + 1) * 40 + n] : 0.f;
      acc  = wmma4(a, bb, acc);
    }
    #pragma unroll
    for (int i = 0; i < 8; ++i)
      if (n < 48) xd[wv][i + half * 8][n] = acc[i];
  }
  __syncthreads();

  // dt = softplus(x_dbl[:, :8] @ dt_proj_w(8x256) + dt_proj_b)
  for (int nt = 0; nt < DI / 16; ++nt) {
    int n = nt * 16 + row;
    v8f acc = {};
    #pragma unroll
    for (int kk = 0; kk < DTR / 4; ++kk) {
      int k = kk * 4 + kb;
      v2f a, bb;
      a.x  = xd[wv][row][k];
      a.y  = xd[wv][row][k + 1];
      bb.x = dt_proj_w[k * DI + n];
      bb.y = dt_proj_w[(k + 1) * DI + n];
      acc  = wmma4(a, bb, acc);
    }
    float bias = dt_proj_b[n];
    size_t base = ((size_t)b * LSEQ + l0) * DI + n;
    #pragma unroll
    for (int i = 0; i < 8; ++i) {
      float v  = acc[i] + bias;
      float sp = (v > 20.f) ? v : log1pf(__expf(v));
      dt[base + (size_t)(i + half * 8) * DI] = sp;
    }
  }
  // scatter B, C (16x16 each)
  for (int t = lane; t < 256; t += 32) {
    int r = t >> 4, n = t & 15;
    size_t o = ((size_t)b * LSEQ + l0 + r) * DS + n;
    Bm[o] = xd[wv][r][DTR + n];
    Cm[o] = xd[wv][r][DTR + DS + n];
  }
}

// =====================================================================
// Chunked selective scan (associative: h_t = a_t*h_{t-1} + b_t).
// k4a: per-chunk local state S and decay product P (h0 = 0).
// k4b: sequential combine over chunks; writes each chunk's INCOMING state
//      in place over S.
// k4c: re-scan chunks seeded with incoming state; emit y over dt (in place).
// Parallelism: B*NCHUNK*DI = 65536 lanes (2048 waves) vs 32 waves before;
// sequential depth 4096 -> ~192 at 2x exp cost (scan was latency-bound).
// =====================================================================
__global__ __launch_bounds__(256) void k4a_chunk_local(
    const float* __restrict__ u, const float* __restrict__ A_log,
    const float* __restrict__ Bm, const float* __restrict__ dt,
    float* __restrict__ Sb, float* __restrict__ Pb) {
  const int bc = blockIdx.x;              // b*NCHUNK + chunk
  const int b  = bc >> 6;                 // / NCHUNK
  const int c  = bc & (NCHUNK - 1);
  const int d  = threadIdx.x;
  float An[DS], h[DS], p[DS];
  #pragma unroll
  for (int n = 0; n < DS; ++n) {
    An[n] = -__expf(A_log[d * DS + n]);
    h[n]  = 0.f;
    p[n]  = 1.f;
  }
  const size_t t0 = (size_t)b * LSEQ + (size_t)c * CLEN;
  for (int i = 0; i < CLEN; ++i) {
    size_t t = t0 + i;
    size_t o = t * DI + d;
    float dtv = dt[o];
    float uv  = u[o];
    if (i + 1 < CLEN) {
      __builtin_prefetch(&dt[o + DI], 0, 1);
      __builtin_prefetch(&u[o + DI], 0, 1);
    }
    const float4* Bp = (const float4*)(Bm + t * DS);
    #pragma unroll
    for (int q = 0; q < 4; ++q) {
      float4 bq = Bp[q];
      float bArr[4] = {bq.x, bq.y, bq.z, bq.w};
      #pragma unroll
      for (int j = 0; j < 4; ++j) {
        int n = q * 4 + j;
        float da = __expf(dtv * An[n]);
        h[n] = fmaf(da, h[n], dtv * bArr[j] * uv);
        p[n] *= da;
      }
    }
  }
  float4* Sp = (float4*)(Sb + ((size_t)bc * DI + d) * DS);
  float4* Pp = (float4*)(Pb + ((size_t)bc * DI + d) * DS);
  #pragma unroll
  for (int q = 0; q < 4; ++q) {
    Sp[q] = make_float4(h[q*4], h[q*4+1], h[q*4+2], h[q*4+3]);
    Pp[q] = make_float4(p[q*4], p[q*4+1], p[q*4+2], p[q*4+3]);
  }
}

__global__ __launch_bounds__(256) void k4b_combine(
    float* __restrict__ Sb, const float* __restrict__ Pb) {
  const int b = blockIdx.x;
  const int d = threadIdx.x;
  float H[DS];
  #pragma unroll
  for (int n = 0; n < DS; ++n) H[n] = 0.f;
  for (int c = 0; c < NCHUNK; ++c) {
    size_t so = (((size_t)b * NCHUNK + c) * DI + d) * DS;
    float4* Sp = (float4*)(Sb + so);
    const float4* Pp = (const float4*)(Pb + so);
    #pragma unroll
    for (int q = 0; q < 4; ++q) {
      float4 s = Sp[q];
      float4 p = Pp[q];
      // write incoming state over S, then advance H = P*H + S
      float4 hin = make_float4(H[q*4], H[q*4+1], H[q*4+2], H[q*4+3]);
      Sp[q] = hin;
      H[q*4]   = fmaf(p.x, hin.x, s.x);
      H[q*4+1] = fmaf(p.y, hin.y, s.y);
      H[q*4+2] = fmaf(p.z, hin.z, s.z);
      H[q*4+3] = fmaf(p.w, hin.w, s.w);
    }
  }
}

__global__ __launch_bounds__(256) void k4c_chunk_emit(
    const float* __restrict__ u, const float* __restrict__ A_log,
    const float* __restrict__ Bm, const float* __restrict__ Cm,
    const float* __restrict__ Sb, float* __restrict__ dt_ys) {
  const int bc = blockIdx.x;
  const int b  = bc >> 6;
  const int c  = bc & (NCHUNK - 1);
  const int d  = threadIdx.x;
  float An[DS], h[DS];
  const float4* Hp = (const float4*)(Sb + ((size_t)bc * DI + d) * DS);
  #pragma unroll
  for (int q = 0; q < 4; ++q) {
    float4 hq = Hp[q];
    h[q*4] = hq.x; h[q*4+1] = hq.y; h[q*4+2] = hq.z; h[q*4+3] = hq.w;
  }
  #pragma unroll
  for (int n = 0; n < DS; ++n) An[n] = -__expf(A_log[d * DS + n]);
  const size_t t0 = (size_t)b * LSEQ + (size_t)c * CLEN;
  for (int i = 0; i < CLEN; ++i) {
    size_t t = t0 + i;
    size_t o = t * DI + d;
    float dtv = dt_ys[o];
    float uv  = u[o];
    if (i + 1 < CLEN) {
      __builtin_prefetch(&dt_ys[o + DI], 0, 1);
      __builtin_prefetch(&u[o + DI], 0, 1);
    }
    const float4* Bp = (const float4*)(Bm + t * DS);
    const float4* Cp = (const float4*)(Cm + t * DS);
    float y = 0.f;
    #pragma unroll
    for (int q = 0; q < 4; ++q) {
      float4 bq = Bp[q], cq = Cp[q];
      float bArr[4] = {bq.x, bq.y, bq.z, bq.w};
      float cArr[4] = {cq.x, cq.y, cq.z, cq.w};
      #pragma unroll
      for (int j = 0; j < 4; ++j) {
        int n = q * 4 + j;
        float da = __expf(dtv * An[n]);
        h[n] = fmaf(da, h[n], dtv * bArr[j] * uv);
        y    = fmaf(h[n], cArr[j], y);
      }
    }
    dt_ys[o] = y;   // same-thread read-then-write: safe in place
  }
}

// =====================================================================
// k5: gate + out_proj (DI->DM) + proj_out (DM->C) + bias + residual
// =====================================================================
#define YG_PAD 260
#define T1_PAD 132
__global__ __launch_bounds__(64) void k5_out(
    const float* __restrict__ ys, const float* __restrict__ u,
    const float* __restrict__ z, const float* __restrict__ Dp,
    const float* __restrict__ out_proj_w, const float* __restrict__ proj_out_w,
    const float* __restrict__ proj_out_b, const float* __restrict__ x,
    float* __restrict__ out) {
  __shared__ float yg[2][16][YG_PAD];
  __shared__ float t1[2][16][T1_PAD];
  const int lane = threadIdx.x & 31;
  const int wv   = threadIdx.x >> 5;
  const int tile = blockIdx.x * 2 + wv;
  const int tok0 = tile * 16;
  const int b    = tok0 >> 12;
  const int l0   = tok0 & (LSEQ - 1);
  const int row  = lane & 15;
  const int half = lane >> 4;
  const int kb   = half * 2;

  // gated activation: (ys + u*D) * silu(z)
  #pragma unroll 4
  for (int it = 0; it < (16 * DI) / 32; ++it) {
    int idx = it * 32 + lane;
    int r = idx >> 8, k = idx & (DI - 1);
    size_t o = ((size_t)b * LSEQ + l0 + r) * DI + k;
    float yv = fmaf(u[o], Dp[k], ys[o]);
    float zv = z[o];
    yg[wv][r][k] = yv * silu_f(zv);
  }
  __syncthreads();

  // t1 = yg(16x256) @ out_proj_w(256x128)
  for (int nt = 0; nt < DM / 16; ++nt) {
    int n = nt * 16 + row;
    v8f acc = {};
    #pragma unroll 4
    for (int kk = 0; kk < DI / 4; ++kk) {
      int k = kk * 4 + kb;
      v2f a, bb;
      a.x  = yg[wv][row][k];
      a.y  = yg[wv][row][k + 1];
      bb.x = out_proj_w[k * DM + n];
      bb.y = out_proj_w[(k + 1) * DM + n];
      acc  = wmma4(a, bb, acc);
    }
    #pragma unroll
    for (int i = 0; i < 8; ++i)
      t1[wv][i + half * 8][n] = acc[i];
  }
  __syncthreads();

  // out = t1(16x128) @ proj_out_w(128x64) + proj_out_b + residual(x)
  for (int nt = 0; nt < CCH / 16; ++nt) {
    int c = nt * 16 + row;
    v8f acc = {};
    #pragma unroll 4
    for (int kk = 0; kk < DM / 4; ++kk) {
      int k = kk * 4 + kb;
      v2f a, bb;
      a.x  = t1[wv][row][k];
      a.y  = t1[wv][row][k + 1];
      bb.x = proj_out_w[k * CCH + c];
      bb.y = proj_out_w[(k + 1) * CCH + c];
      acc  = wmma4(a, bb, acc);
    }
    float bias = proj_out_b[c];
    const float* xb = x   + ((size_t)b * CCH + c) * LSEQ + l0;
    float*       ob = (float*)out + ((size_t)b * CCH + c) * LSEQ + l0;
    #pragma unroll
    for (int i = 0; i < 8; ++i) {
      int r = i + half * 8;
      ob[r] = acc[i] + bias + xb[r];
    }
  }
}

// =====================================================================
extern "C" void kernel_launch(void* const* d_in, const int* in_sizes, int n_in,
                              void* d_out, int out_size, void* d_ws, size_t ws_size,
                              hipStream_t stream) {
  const float* x          = (const float*)d_in[0];
  const float* proj_w     = (const float*)d_in[1];
  const float* proj_b     = (const float*)d_in[2];
  const float* ln_g       = (const float*)d_in[3];
  const float* ln_b       = (const float*)d_in[4];
  const float* in_proj_w  = (const float*)d_in[5];
  const float* conv_w     = (const float*)d_in[6];
  const float* conv_b     = (const float*)d_in[7];
  const float* x_proj_w   = (const float*)d_in[8];
  const float* dt_proj_w  = (const float*)d_in[9];
  const float* dt_proj_b  = (const float*)d_in[10];
  const float* A_log      = (const float*)d_in[11];
  const float* Dp         = (const float*)d_in[12];
  const float* out_proj_w = (const float*)d_in[13];
  const float* proj_out_w = (const float*)d_in[14];
  const float* proj_out_b = (const float*)d_in[15];

  // workspace layout (floats); W0 is reused: xi -> dt -> ys (each transition
  // is a full overwrite or an in-place same-thread read-then-write).
  float* ws = (float*)d_ws;
  const size_t BLD = (size_t)BATCH * LSEQ * DI;      // 4,194,304
  const size_t BLS = (size_t)BATCH * LSEQ * DS;      //   262,144
  const size_t CHS = (size_t)BATCH * NCHUNK * DI * DS; // 1,048,576
  float* W0 = ws;                 // xi / dt / ys
  float* W1 = ws + BLD;           // z
  float* W2 = ws + 2 * BLD;       // u
  float* Bm = ws + 3 * BLD;       // (B,L,16)
  float* Cm = Bm + BLS;
  float* Sb = Cm + BLS;           // chunk states  (B,NCHUNK,DI,DS)
  float* Pb = Sb + CHS;           // chunk decay products

  const int nTiles = BATCH * LSEQ / 16;              // 1024
  const int nBlk   = nTiles / 2;                     // 2 wave-tiles per block

  k1_proj_ln_inproj<<<nBlk, 64, 0, stream>>>(x, proj_w, proj_b, ln_g, ln_b,
                                             in_proj_w, W0, W1);
  k2_conv_silu<<<(BATCH * LSEQ * DI) / 256, 256, 0, stream>>>(W0, conv_w,
                                                              conv_b, W2);
  k3_xproj_dt<<<nBlk, 64, 0, stream>>>(W2, x_proj_w, dt_proj_w, dt_proj_b,
                                       W0, Bm, Cm);
  k4a_chunk_local<<<BATCH * NCHUNK, DI, 0, stream>>>(W2, A_log, Bm, W0, Sb, Pb);
  k4b_combine<<<BATCH, DI, 0, stream>>>(Sb, Pb);
  k4c_chunk_emit<<<BATCH * NCHUNK, DI, 0, stream>>>(W2, A_log, Bm, Cm, Sb, W0);
  k5_out<<<nBlk, 64, 0, stream>>>(W0, W2, W1, Dp, out_proj_w, proj_out_w,
                                  proj_out_b, x, (float*)d_out);
}